// HadamardClassifier_44487271252814
// MI455X (gfx1250) — compile-verified
//
#include <hip/hip_runtime.h>

// ---------------------------------------------------------------------------
// HadamardClassifier for MI455X (gfx1250, wave32)
//
// out[b, j] = -scale * sum_i xn[b,i] * (-1)^popcount(i & j) + bias[j]
// Since i < 2048, popcount(i&j) only sees the low 11 bits of j, so
//   out[b, j] = -scale * FWHT2048(xn[b])[j & 2047] + bias[j].
// FWHT2048 = H16 (x) H128:
//   - H128 on the low 7 index bits: LDS butterfly stages (adds/subs).
//   - H16 on the high 4 index bits: V_WMMA_F32_16X16X4_F32, 4 chained
//     K=4 steps per 16x16 tile, one tile per wave (8 waves x 16 cols = 128).
// Bound by the 245 MB output stream (~12 us at 23.3 TB/s): output uses
// non-temporal b128 stores so the L2 keeps bias resident; x uses NT loads.
// ---------------------------------------------------------------------------

typedef __attribute__((ext_vector_type(2))) float v2f;
typedef __attribute__((ext_vector_type(4))) float v4f;
typedef __attribute__((ext_vector_type(8))) float v8f;

#define IN_DIM  2048
#define OUT_DIM 14951
#define BLOCK   256

__device__ __forceinline__ float hsign(int m, int k) {
    // Sylvester Hadamard entry H[m][k] = (-1)^popcount(m & k)
    return (__popc(m & k) & 1) ? -1.0f : 1.0f;
}

__global__ __launch_bounds__(BLOCK)
void HadamardClassifier_44487271252814_kernel(const float* __restrict__ x,
                                              const float* __restrict__ scale,
                                              const float* __restrict__ bias,
                                              float* __restrict__ out)
{
    __shared__ __align__(16) float W[IN_DIM];   // row workspace, viewed as [16][128]
    __shared__ float red[BLOCK];

    const int t = threadIdx.x;
    const int b = blockIdx.x;

    // Warm the bias stream into cache (global_prefetch_b8 path).
    for (int j = t * 64; j < OUT_DIM; j += BLOCK * 64)
        __builtin_prefetch(bias + j, 0, 1);

    // ---- Load row (2048 f32 = 512 float4) + sum of squares -----------------
    // Each row is read exactly once across the dispatch -> non-temporal loads.
    const v4f* row4 = (const v4f*)(x + (size_t)b * IN_DIM);
    v4f a0 = __builtin_nontemporal_load(row4 + t);
    v4f a1 = __builtin_nontemporal_load(row4 + t + BLOCK);
    float ss = a0.x * a0.x + a0.y * a0.y + a0.z * a0.z + a0.w * a0.w
             + a1.x * a1.x + a1.y * a1.y + a1.z * a1.z + a1.w * a1.w;
    red[t] = ss;
    __syncthreads();
#pragma unroll
    for (int s = BLOCK / 2; s > 0; s >>= 1) {
        if (t < s) red[t] += red[t + s];
        __syncthreads();
    }
    const float inv = rsqrtf(fmaxf(red[0], 1e-12f));  // l2_normalize epsilon

    v4f* W4 = (v4f*)W;
    W4[t]         = a0 * inv;
    W4[t + BLOCK] = a1 * inv;
    __syncthreads();

    // ---- H128 over the low 7 index bits: 7 butterfly stages in LDS ---------
#pragma unroll
    for (int s = 0; s < 7; ++s) {
        const int h = 1 << s;
#pragma unroll
        for (int p = 0; p < 4; ++p) {        // 1024 pairs / 256 threads
            const int pr = t + p * BLOCK;
            const int i  = ((pr >> s) << (s + 1)) | (pr & (h - 1));
            const float a = W[i];
            const float c = W[i + h];
            W[i]     = a + c;
            W[i + h] = a - c;
        }
        __syncthreads();
    }

    // ---- H16 over the high 4 bits: T = H16 * Y via V_WMMA_F32_16X16X4_F32 --
    // Y = W viewed as [i1=16][s=128]; each wave owns one 16-column tile.
    {
        const int lane    = t & 31;
        const int wave    = t >> 5;          // 0..7
        const int colBase = wave * 16;
        const int half    = lane >> 4;       // 0: lanes 0-15, 1: lanes 16-31
        const int l15     = lane & 15;

        v8f acc = {};
#pragma unroll
        for (int kb = 0; kb < 4; ++kb) {
            // A (16x4 f32): lanes 0-15 hold K={4kb+0,4kb+1}, lanes 16-31 K={4kb+2,4kb+3}
            const int k0 = kb * 4 + 2 * half;
            v2f A, B;
            A.x = hsign(l15, k0);
            A.y = hsign(l15, k0 + 1);
            // B (4x16 f32): VGPR0 rows {K0|K2}, VGPR1 rows {K1|K3} across lane halves
            B.x = W[(k0 + 0) * 128 + colBase + l15];
            B.y = W[(k0 + 1) * 128 + colBase + l15];
            acc = __builtin_amdgcn_wmma_f32_16x16x4_f32(
                /*neg_a=*/false, A, /*neg_b=*/false, B,
                /*c_mod=*/(short)0, acc, /*reuse_a=*/false, /*reuse_b=*/false);
        }
        // C/D layout: VGPR v -> row v (lanes 0-15), row v+8 (lanes 16-31)
#pragma unroll
        for (int v = 0; v < 8; ++v) {
            const int m = v + 8 * half;
            W[m * 128 + colBase + l15] = acc[v];
        }
    }
    __syncthreads();

    // ---- Stream output: out[b,j] = -scale * W[j & 2047] + bias[j] ----------
    // Output is written once and never re-read -> non-temporal b128 stores.
    // Row base b*14951 is not 16B-aligned for every b: peel a scalar prologue
    // up to 16B alignment, stream aligned float4s, then a scalar epilogue.
    const float sc = -scale[0];
    const size_t base = (size_t)b * OUT_DIM;
    float* orow = out + base;

    const int pre = (int)((4 - (base & 3)) & 3);       // 0..3 scalar elements
    const int nq  = (OUT_DIM - pre) >> 2;              // aligned float4 count
    const int rem = (OUT_DIM - pre) & 3;               // 0..3 tail elements

    if (t < pre)
        orow[t] = fmaf(sc, W[t & 2047], bias[t]);

    v4f* ov = (v4f*)(orow + pre);
    for (int q = t; q < nq; q += BLOCK) {
        const int j = pre + 4 * q;
        v4f v;
        v.x = fmaf(sc, W[(j + 0) & 2047], bias[j + 0]);
        v.y = fmaf(sc, W[(j + 1) & 2047], bias[j + 1]);
        v.z = fmaf(sc, W[(j + 2) & 2047], bias[j + 2]);
        v.w = fmaf(sc, W[(j + 3) & 2047], bias[j + 3]);
        __builtin_nontemporal_store(v, ov + q);
    }

    if (t < rem) {
        const int j = OUT_DIM - rem + t;
        orow[j] = fmaf(sc, W[j & 2047], bias[j]);
    }
}

extern "C" void kernel_launch(void* const* d_in, const int* in_sizes, int n_in,
                              void* d_out, int out_size, void* d_ws, size_t ws_size,
                              hipStream_t stream) {
    (void)in_sizes; (void)n_in; (void)d_ws; (void)ws_size;
    const float* x     = (const float*)d_in[0];
    const float* scale = (const float*)d_in[1];
    const float* bias  = (const float*)d_in[2];
    float* out         = (float*)d_out;

    const int batch = 4096;                  // BATCH rows, one block per row
    HadamardClassifier_44487271252814_kernel<<<batch, BLOCK, 0, stream>>>(
        x, scale, bias, out);
    (void)out_size;
}